// HypernymVisual_56710748176755
// MI455X (gfx1250) — compile-verified
//
#include <hip/hip_runtime.h>
#include <hip/hip_bf16.h>

typedef __attribute__((ext_vector_type(2))) float v2f;
typedef __attribute__((ext_vector_type(8))) float v8f;

#define D_V 4096
#define D_H 1024
#define D_E 100
#define D_EP 112   // D_E padded to 7*16 for WMMA tiling
#define N_ROI 256
#define L_LAB 20000

// ---------------------------------------------------------------------------
// Kernel 1: h = relu(vfs) @ W1 + b1     [256 x 4096] @ [4096 x 1024]
// One wave per 64x16 output block (4 stacked 16x16 WMMA tiles along M) so the
// B fragment is loaded once and reused 4x. f32 WMMA, K stepped by 4.
// A frag (16x4):  lane holds A[lane%16][2*(lane/16)+v], v in {0,1}
// B frag (4x16):  lane holds B[2*(lane/16)+v][lane%16]
// C/D (16x16):    vgpr r: lanes 0-15 -> M=r, lanes 16-31 -> M=r+8; N=lane%16
// ---------------------------------------------------------------------------
__global__ __launch_bounds__(32) void gemm1_relu_wmma(
    const float* __restrict__ vfs, const float* __restrict__ W1,
    const float* __restrict__ b1, float* __restrict__ h)
{
    const int tileN = blockIdx.x & 63;        // 64 tiles along D_H
    const int mblk  = blockIdx.x >> 6;        // 4 blocks of 64 rows along N_ROI
    const int lane  = threadIdx.x & 31;
    const int lm    = lane & 15;
    const int khalf = (lane >> 4) << 1;       // 0 or 2
    const int col   = tileN * 16 + lm;        // N for B loads / stores

    const float* __restrict__ ap0 = vfs + (size_t)(mblk * 64 + lm) * D_V;

    v8f acc[4] = {};
#pragma unroll 4
    for (int k0 = 0; k0 < D_V; k0 += 4) {
        v2f b;
        b.x = W1[(size_t)(k0 + khalf + 0) * D_H + col];
        b.y = W1[(size_t)(k0 + khalf + 1) * D_H + col];
#pragma unroll
        for (int t = 0; t < 4; ++t) {         // 4 M-tiles share this B frag
            const float* __restrict__ ap = ap0 + (size_t)t * 16 * D_V;
            v2f a;
            a.x = fmaxf(ap[k0 + khalf + 0], 0.0f);   // fused ReLU on A
            a.y = fmaxf(ap[k0 + khalf + 1], 0.0f);
            acc[t] = __builtin_amdgcn_wmma_f32_16x16x4_f32(
                false, a, false, b, (short)0, acc[t], false, false);
        }
    }

    const float bias = b1[col];
    const int mtop = (lane >> 4) << 3;        // 0 or 8
#pragma unroll
    for (int t = 0; t < 4; ++t) {
#pragma unroll
        for (int r = 0; r < 8; ++r) {
            const int m = mblk * 64 + t * 16 + mtop + r;
            h[(size_t)m * D_H + col] = acc[t][r] + bias;
        }
    }
}

// ---------------------------------------------------------------------------
// Kernel 2: e = relu(h) @ W2 + b2       [256 x 1024] @ [1024 x 100]
// N padded to 112; e stored with stride D_EP in workspace.
// ---------------------------------------------------------------------------
__global__ __launch_bounds__(32) void gemm2_relu_wmma(
    const float* __restrict__ h, const float* __restrict__ W2,
    const float* __restrict__ b2, float* __restrict__ e)
{
    const int tileN = blockIdx.x % 7;         // 7 tiles cover 112 (100 valid)
    const int tileM = blockIdx.x / 7;         // 16 tiles along N_ROI
    const int lane  = threadIdx.x & 31;
    const int lm    = lane & 15;
    const int khalf = (lane >> 4) << 1;
    const int arow  = tileM * 16 + lm;
    const int col   = tileN * 16 + lm;
    const bool colv = (col < D_E);

    const float* __restrict__ ap = h + (size_t)arow * D_H;

    v8f acc = {};
#pragma unroll 4
    for (int k0 = 0; k0 < D_H; k0 += 4) {
        v2f a, b;
        a.x = fmaxf(ap[k0 + khalf + 0], 0.0f);
        a.y = fmaxf(ap[k0 + khalf + 1], 0.0f);
        b.x = colv ? W2[(size_t)(k0 + khalf + 0) * D_E + col] : 0.0f;
        b.y = colv ? W2[(size_t)(k0 + khalf + 1) * D_E + col] : 0.0f;
        acc = __builtin_amdgcn_wmma_f32_16x16x4_f32(
            false, a, false, b, (short)0, acc, false, false);
    }

    const float bias = colv ? b2[col] : 0.0f;
    const int mtop = (lane >> 4) << 3;
#pragma unroll
    for (int r = 0; r < 8; ++r) {
        const int m = tileM * 16 + mtop + r;
        e[(size_t)m * D_EP + col] = acc[r] + bias;
    }
}

// ---------------------------------------------------------------------------
// Kernel 3: scores[n][l] = -sqrt( sum_d relu(lv[l][d] - e[n][d])^2 )
// Non-bilinear (relu before square) -> VALU. Each thread owns one l and keeps
// its label vector in 100 registers; all 256 n are swept with the e tile
// staged through LDS via gfx1250 async global->LDS loads (ASYNCcnt-tracked,
// no VGPR round-trip). label_vecs read exactly once.
// ---------------------------------------------------------------------------
__global__ __launch_bounds__(256) void order_scores(
    const float* __restrict__ label_vecs, const float* __restrict__ e,
    float* __restrict__ out)
{
    __shared__ float es[16][D_E];

    const int l = blockIdx.x * 256 + threadIdx.x;
    const bool lvalid = (l < L_LAB);

    float lv[D_E];
#pragma unroll
    for (int d = 0; d < D_E; ++d)
        lv[d] = lvalid ? label_vecs[(size_t)l * D_E + d] : 0.0f;

    for (int n0 = 0; n0 < N_ROI; n0 += 16) {
        __syncthreads();                      // previous tile fully consumed

        // Stage e[n0:n0+16, 0:100] into LDS with async global->LDS copies.
        for (int i = threadIdx.x; i < 16 * D_E; i += 256) {
            const int nn = i / D_E;
            const int dd = i % D_E;
            // LDS byte address = generic shared pointer truncated to 32 bits
            // (flat->LDS mapping uses addr[31:0], ISA 10.2).
            unsigned lds_addr = (unsigned)(size_t)(&es[nn][dd]);
            unsigned long long gaddr =
                (unsigned long long)(size_t)(&e[(size_t)(n0 + nn) * D_EP + dd]);
            asm volatile("global_load_async_to_lds_b32 %0, %1, off"
                         :: "v"(lds_addr), "v"(gaddr)
                         : "memory");
        }
        asm volatile("s_wait_asynccnt 0x0" ::: "memory");
        __syncthreads();

        if (lvalid) {
            for (int nn = 0; nn < 16; ++nn) {
                float s = 0.0f;
#pragma unroll
                for (int d = 0; d < D_E; ++d) {
                    float diff = lv[d] - es[nn][d];
                    diff = fmaxf(diff, 0.0f);
                    s = fmaf(diff, diff, s);
                }
                out[(size_t)(n0 + nn) * L_LAB + l] = -sqrtf(s);
            }
        }
    }
}

// ---------------------------------------------------------------------------
extern "C" void kernel_launch(void* const* d_in, const int* in_sizes, int n_in,
                              void* d_out, int out_size, void* d_ws, size_t ws_size,
                              hipStream_t stream) {
    const float* vfs = (const float*)d_in[0];   // [256, 4096]
    const float* lv  = (const float*)d_in[1];   // [20000, 100]
    const float* W1  = (const float*)d_in[2];   // [4096, 1024]
    const float* b1  = (const float*)d_in[3];   // [1024]
    const float* W2  = (const float*)d_in[4];   // [1024, 100]
    const float* b2  = (const float*)d_in[5];   // [100]
    float* out = (float*)d_out;                 // [256, 20000]

    float* h = (float*)d_ws;                    // 256*1024 f32 = 1 MB
    float* e = h + (size_t)N_ROI * D_H;         // 256*112  f32

    gemm1_relu_wmma<<<4 * 64, 32, 0, stream>>>(vfs, W1, b1, h);
    gemm2_relu_wmma<<<16 * 7, 32, 0, stream>>>(h, W2, b2, e);
    order_scores<<<(L_LAB + 255) / 256, 256, 0, stream>>>(lv, e, out);
}